// DynamicGAT_25056839205718
// MI455X (gfx1250) — compile-verified
//
#include <hip/hip_runtime.h>
#include <hip/hip_bf16.h>

// ---------------------------------------------------------------------------
// DynamicGAT forward for MI455X (gfx1250), wave32, f32 WMMA 16x16x4.
//   N=4096 nodes, D=256 features, NHID=64, NHEADS=4, OUT=2, K=5, ALPHA=0.2
// Dominant cost: 4096x4096x256 Gram matrix for kNN -> fused WMMA + top-6.
// ---------------------------------------------------------------------------

#define GAT_N      4096
#define GAT_D      256
#define GAT_NHID   64
#define GAT_NHEADS 4
#define GAT_OUT    2
#define GAT_KP1    6          // K + 1 neighbours (includes self)
#define GAT_ALPHA  0.2f
#define GAT_LNEPS  1e-5f

typedef __attribute__((ext_vector_type(2))) float v2f;
typedef __attribute__((ext_vector_type(8))) float v8f;

// D(16x16,f32) = A(16x4,f32) * B(4x16,f32) + C
// A frag: lane(l&15)=row M, vgpr0 -> K = base + (l>>4)*2, vgpr1 -> K+1
// B frag: lane(l&15)=col N, vgpr0 -> K = base + (l>>4)*2, vgpr1 -> K+1
// C/D   : vgpr r -> row r + (l>>4)*8, col = l&15
__device__ __forceinline__ v8f wmma4(v2f a, v2f b, v8f c) {
  return __builtin_amdgcn_wmma_f32_16x16x4_f32(
      /*neg_a=*/false, a, /*neg_b=*/false, b,
      /*c_mod=*/(short)0, c, /*reuse_a=*/false, /*reuse_b=*/false);
}

// ---------------------------------------------------------------------------
// Generic fp32 WMMA GEMM: C[M x Nc] = A[M x Kd] * B[Kd x Nc] + bias
// One wave per 16x16 output tile. Grid must supply exactly M/16 * Nc/16 waves.
// ---------------------------------------------------------------------------
__global__ __launch_bounds__(128) void gemm_wmma_bias_kernel(
    const float* __restrict__ A, const float* __restrict__ B,
    const float* __restrict__ bias, float* __restrict__ C,
    int M, int Nc, int Kd, int ldc)
{
  const int lane  = threadIdx.x & 31;
  const int wave  = threadIdx.x >> 5;
  const int half  = lane >> 4;
  const int ln    = lane & 15;
  const int tile  = blockIdx.x * (blockDim.x >> 5) + wave;
  const int tilesN  = Nc >> 4;
  const int rowTile = tile / tilesN;
  const int colTile = tile - rowTile * tilesN;

  const int aRow = rowTile * 16 + ln;
  const int bCol = colTile * 16 + ln;
  const float* __restrict__ arow = A + (size_t)aRow * Kd;

  v8f acc = {};
  for (int kk = 0; kk < Kd; kk += 4) {
    const int k0 = kk + half * 2;
    v2f av = *(const v2f*)(arow + k0);            // 8B-aligned (k0 even)
    v2f bv;
    bv.x = B[(size_t)k0       * Nc + bCol];
    bv.y = B[(size_t)(k0 + 1) * Nc + bCol];
    acc = wmma4(av, bv, acc);
  }

  const float bb = bias ? bias[bCol] : 0.0f;
#pragma unroll
  for (int r = 0; r < 8; ++r) {
    const int row = rowTile * 16 + half * 8 + r;
    C[(size_t)row * ldc + colTile * 16 + ln] = acc[r] + bb;
  }
}

// ---------------------------------------------------------------------------
// sq[i] = sum_j xm[i][j]^2 ; one wave per row
// ---------------------------------------------------------------------------
__global__ __launch_bounds__(256) void rowsq_kernel(
    const float* __restrict__ xm, float* __restrict__ sq, int Nn, int Dd)
{
  const int row  = (blockIdx.x * blockDim.x + threadIdx.x) >> 5;
  const int lane = threadIdx.x & 31;
  if (row >= Nn) return;
  const float* __restrict__ r = xm + (size_t)row * Dd;
  float s = 0.0f;
  for (int j = lane; j < Dd; j += 32) { float v = r[j]; s += v * v; }
#pragma unroll
  for (int off = 16; off > 0; off >>= 1) s += __shfl_xor(s, off, 32);
  if (lane == 0) sq[row] = s;
}

// ---------------------------------------------------------------------------
// Fused Gram + dist2 + top-6 kNN. One wave per 16-row strip.
// Per outer iteration: 16x64 distance block via 4 independent WMMA chains
// (K=256, step 4), staged through LDS; lanes 0..15 keep a running top-6
// (smallest dist2) per row in registers. Set-equality with jax top_k.
// ---------------------------------------------------------------------------
__global__ __launch_bounds__(32) void knn_topk_kernel(
    const float* __restrict__ xm, const float* __restrict__ sq,
    int* __restrict__ nnidx, int Nn, int Dd)
{
  __shared__ float lds[16][68];                   // +4 pad: conflict-free
  const int lane = threadIdx.x & 31;
  const int half = lane >> 4;
  const int ln   = lane & 15;
  const int rowBase = blockIdx.x << 4;
  const float* __restrict__ arow = xm + (size_t)(rowBase + ln) * Dd;

  float sqr[8];
#pragma unroll
  for (int r = 0; r < 8; ++r) sqr[r] = sq[rowBase + half * 8 + r];

  float best[GAT_KP1];
  int   bidx[GAT_KP1];
#pragma unroll
  for (int t = 0; t < GAT_KP1; ++t) { best[t] = 3.0e38f; bidx[t] = 0; }

  for (int cb = 0; cb < Nn; cb += 64) {
    const float* __restrict__ b0 = xm + (size_t)(cb      + ln) * Dd;
    const float* __restrict__ b1 = xm + (size_t)(cb + 16 + ln) * Dd;
    const float* __restrict__ b2 = xm + (size_t)(cb + 32 + ln) * Dd;
    const float* __restrict__ b3 = xm + (size_t)(cb + 48 + ln) * Dd;
    v8f acc0 = {}, acc1 = {}, acc2 = {}, acc3 = {};
    for (int kk = 0; kk < Dd; kk += 4) {
      const int k0 = kk + half * 2;
      v2f av = *(const v2f*)(arow + k0);
      acc0 = wmma4(av, *(const v2f*)(b0 + k0), acc0);
      acc1 = wmma4(av, *(const v2f*)(b1 + k0), acc1);
      acc2 = wmma4(av, *(const v2f*)(b2 + k0), acc2);
      acc3 = wmma4(av, *(const v2f*)(b3 + k0), acc3);
    }
    const float sqc0 = sq[cb      + ln];
    const float sqc1 = sq[cb + 16 + ln];
    const float sqc2 = sq[cb + 32 + ln];
    const float sqc3 = sq[cb + 48 + ln];
#pragma unroll
    for (int r = 0; r < 8; ++r) {
      const int rr = half * 8 + r;
      lds[rr][ 0 + ln] = sqr[r] + sqc0 - 2.0f * acc0[r];
      lds[rr][16 + ln] = sqr[r] + sqc1 - 2.0f * acc1[r];
      lds[rr][32 + ln] = sqr[r] + sqc2 - 2.0f * acc2[r];
      lds[rr][48 + ln] = sqr[r] + sqc3 - 2.0f * acc3[r];
    }
    __syncthreads();
    if (lane < 16) {
      for (int c = 0; c < 64; ++c) {
        const float d = lds[lane][c];
        const int   j = cb + c;
        float worst = best[0]; int wi = 0;
#pragma unroll
        for (int t = 1; t < GAT_KP1; ++t)
          if (best[t] > worst) { worst = best[t]; wi = t; }
        if (d < worst) { best[wi] = d; bidx[wi] = j; }
      }
    }
    __syncthreads();
  }

  if (lane < 16) {
    const int row = rowBase + lane;
#pragma unroll
    for (int t = 0; t < GAT_KP1; ++t) nnidx[row * GAT_KP1 + t] = bidx[t];
  }
}

// ---------------------------------------------------------------------------
// e1[h,n] = Wh[h,n,:].a1[h,:], e2[h,n] = Wh[h,n,:].a2[h,:] ; wave per (h,n)
// ---------------------------------------------------------------------------
__global__ __launch_bounds__(256) void e12_kernel(
    const float* __restrict__ Wh, const float* __restrict__ avec,
    float* __restrict__ e1, float* __restrict__ e2, int Nn)
{
  const int gw   = (blockIdx.x * blockDim.x + threadIdx.x) >> 5;
  const int lane = threadIdx.x & 31;
  if (gw >= GAT_NHEADS * Nn) return;
  const int hh = gw / Nn;
  const int n  = gw - hh * Nn;
  const float* __restrict__ row = Wh + ((size_t)hh * Nn + n) * GAT_NHID;
  const float* __restrict__ a1  = avec + (size_t)hh * 2 * GAT_NHID;
  const float* __restrict__ a2  = a1 + GAT_NHID;
  float s1 = 0.0f, s2 = 0.0f;
#pragma unroll
  for (int j = lane; j < GAT_NHID; j += 32) {
    const float v = row[j];
    s1 += v * a1[j];
    s2 += v * a2[j];
  }
#pragma unroll
  for (int off = 16; off > 0; off >>= 1) {
    s1 += __shfl_xor(s1, off, 32);
    s2 += __shfl_xor(s2, off, 32);
  }
  if (lane == 0) { e1[(size_t)hh * Nn + n] = s1; e2[(size_t)hh * Nn + n] = s2; }
}

// ---------------------------------------------------------------------------
// Per (h,n): softmax over 6 neighbours of leaky_relu(e1[n]+e2[j]),
// hbuf[n, h*64+d] += sum_t att_t * Wh[h, j_t, d]. Wave per (h,n).
// ---------------------------------------------------------------------------
__global__ __launch_bounds__(256) void agg_kernel(
    const int* __restrict__ nnidx, const float* __restrict__ e1,
    const float* __restrict__ e2, const float* __restrict__ Wh,
    float* __restrict__ hbuf, int Nn)
{
  const int gw   = (blockIdx.x * blockDim.x + threadIdx.x) >> 5;
  const int lane = threadIdx.x & 31;
  if (gw >= GAT_NHEADS * Nn) return;
  const int hh = gw / Nn;
  const int n  = gw - hh * Nn;

  float lg[GAT_KP1];
  int   jj[GAT_KP1];
  const float me = e1[(size_t)hh * Nn + n];
  float mx = -3.0e38f;
#pragma unroll
  for (int t = 0; t < GAT_KP1; ++t) {
    const int j = nnidx[n * GAT_KP1 + t];          // uniform across wave
    jj[t] = j;
    float v = me + e2[(size_t)hh * Nn + j];
    v = (v > 0.0f) ? v : GAT_ALPHA * v;            // leaky relu
    lg[t] = v;
    mx = fmaxf(mx, v);
  }
  float ssum = 0.0f;
#pragma unroll
  for (int t = 0; t < GAT_KP1; ++t) { lg[t] = __expf(lg[t] - mx); ssum += lg[t]; }
  const float inv = 1.0f / ssum;

  float acc0 = 0.0f, acc1 = 0.0f;
#pragma unroll
  for (int t = 0; t < GAT_KP1; ++t) {
    const float* __restrict__ wr = Wh + ((size_t)hh * Nn + jj[t]) * GAT_NHID;
    const float w = lg[t] * inv;
    acc0 += w * wr[lane];
    acc1 += w * wr[lane + 32];
  }
  float* __restrict__ dst = hbuf + (size_t)n * (GAT_NHEADS * GAT_NHID) + hh * GAT_NHID;
  dst[lane]      += acc0;     // disjoint slice per (h,n): no atomics needed
  dst[lane + 32] += acc1;
}

// ---------------------------------------------------------------------------
// Per row: LayerNorm(256) -> ELU -> @ Wo(256x2) + bo. Wave per row.
// ---------------------------------------------------------------------------
__global__ __launch_bounds__(256) void ln_elu_out_kernel(
    const float* __restrict__ hbuf, const float* __restrict__ gam,
    const float* __restrict__ bet, const float* __restrict__ Wo,
    const float* __restrict__ bo, float* __restrict__ out, int Nn)
{
  const int row  = (blockIdx.x * blockDim.x + threadIdx.x) >> 5;
  const int lane = threadIdx.x & 31;
  if (row >= Nn) return;
  const int C = GAT_NHEADS * GAT_NHID;             // 256
  const float* __restrict__ r = hbuf + (size_t)row * C;

  float v[8];
  float s = 0.0f;
#pragma unroll
  for (int i = 0; i < 8; ++i) { v[i] = r[lane + 32 * i]; s += v[i]; }
#pragma unroll
  for (int off = 16; off > 0; off >>= 1) s += __shfl_xor(s, off, 32);
  const float mu = s * (1.0f / C);

  float vs = 0.0f;
#pragma unroll
  for (int i = 0; i < 8; ++i) { const float d = v[i] - mu; vs += d * d; }
#pragma unroll
  for (int off = 16; off > 0; off >>= 1) vs += __shfl_xor(vs, off, 32);
  const float rstd = rsqrtf(vs * (1.0f / C) + GAT_LNEPS);

  float o0 = 0.0f, o1 = 0.0f;
#pragma unroll
  for (int i = 0; i < 8; ++i) {
    const int c = lane + 32 * i;
    float hv = (v[i] - mu) * rstd * gam[c] + bet[c];
    hv = (hv > 0.0f) ? hv : (__expf(hv) - 1.0f);   // elu(alpha=1)
    o0 += hv * Wo[c * GAT_OUT + 0];
    o1 += hv * Wo[c * GAT_OUT + 1];
  }
#pragma unroll
  for (int off = 16; off > 0; off >>= 1) {
    o0 += __shfl_xor(o0, off, 32);
    o1 += __shfl_xor(o1, off, 32);
  }
  if (lane == 0) {
    out[(size_t)row * GAT_OUT + 0] = o0 + bo[0];
    out[(size_t)row * GAT_OUT + 1] = o1 + bo[1];
  }
}

// ---------------------------------------------------------------------------
extern "C" void kernel_launch(void* const* d_in, const int* in_sizes, int n_in,
                              void* d_out, int out_size, void* d_ws, size_t ws_size,
                              hipStream_t stream) {
  (void)in_sizes; (void)n_in; (void)out_size; (void)ws_size;

  const float* x    = (const float*)d_in[0];   // (N, D)
  const float* Wm   = (const float*)d_in[1];   // (D, D)
  const float* bm   = (const float*)d_in[2];   // (D,)
  const float* W    = (const float*)d_in[3];   // (H, D, NHID)
  const float* avec = (const float*)d_in[4];   // (H, 2*NHID, 1)
  const float* Wr   = (const float*)d_in[5];   // (D, H*NHID)
  const float* br   = (const float*)d_in[6];   // (H*NHID,)
  const float* ln_g = (const float*)d_in[7];   // (H*NHID,)
  const float* ln_b = (const float*)d_in[8];   // (H*NHID,)
  const float* Wo   = (const float*)d_in[9];   // (H*NHID, OUT)
  const float* bo   = (const float*)d_in[10];  // (OUT,)
  float* out = (float*)d_out;

  const int N = GAT_N, D = GAT_D;

  // Workspace layout (floats):
  float* xm  = (float*)d_ws;                                  // N*D
  float* sqb = xm  + (size_t)N * D;                           // N
  float* Whb = sqb + N;                                       // H*N*NHID
  float* e1b = Whb + (size_t)GAT_NHEADS * N * GAT_NHID;       // H*N
  float* e2b = e1b + (size_t)GAT_NHEADS * N;                  // H*N
  int*   nnb = (int*)(e2b + (size_t)GAT_NHEADS * N);          // N*6 ints
  float* hb  = (float*)(nnb + (size_t)N * GAT_KP1);           // N*D

  // 1) xm = x @ Wm + bm          (4096x256x256, WMMA f32)
  {
    const int tiles = (N / 16) * (D / 16);                    // 4096 tiles
    gemm_wmma_bias_kernel<<<tiles / 4, 128, 0, stream>>>(x, Wm, bm, xm, N, D, D, D);
  }
  // 2) sq[i] = |xm_i|^2
  rowsq_kernel<<<(N * 32) / 256, 256, 0, stream>>>(xm, sqb, N, D);
  // 3) fused Gram + dist2 + top-6   (4096x4096x256, WMMA f32)
  knn_topk_kernel<<<N / 16, 32, 0, stream>>>(xm, sqb, nnb, N, D);
  // 4) Wh[h] = x @ W[h]          (4 x 4096x256x64, WMMA f32)
  for (int h = 0; h < GAT_NHEADS; ++h) {
    const int tiles = (N / 16) * (GAT_NHID / 16);             // 1024 tiles
    gemm_wmma_bias_kernel<<<tiles / 4, 128, 0, stream>>>(
        x, W + (size_t)h * D * GAT_NHID, nullptr,
        Whb + (size_t)h * N * GAT_NHID, N, GAT_NHID, D, GAT_NHID);
  }
  // 5) attention logit halves e1, e2
  e12_kernel<<<(GAT_NHEADS * N * 32) / 256, 256, 0, stream>>>(Whb, avec, e1b, e2b, N);
  // 6) h = x @ Wr + br           (4096x256x256, WMMA f32)
  {
    const int tiles = (N / 16) * (D / 16);
    gemm_wmma_bias_kernel<<<tiles / 4, 128, 0, stream>>>(x, Wr, br, hb, N, D, D, D);
  }
  // 7) h += softmax-attention aggregation over 6 neighbours
  agg_kernel<<<(GAT_NHEADS * N * 32) / 256, 256, 0, stream>>>(nnb, e1b, e2b, Whb, hb, N);
  // 8) LayerNorm -> ELU -> @Wo + bo
  ln_elu_out_kernel<<<(N * 32) / 256, 256, 0, stream>>>(hb, ln_g, ln_b, Wo, bo, out, N);
}